// HashEncoding_11184094839334
// MI455X (gfx1250) — compile-verified
//
#include <hip/hip_runtime.h>
#include <hip/hip_bf16.h>
#include <stdint.h>

// ---------------------------------------------------------------------------
// Multi-resolution hash encoding (no interpolation variant).
//   out[n, 2l:2l+2] = tables[l][ hash(trunc(xyz[n] * 16*2^l)) ]
//   hash(x,y,z) = (x*1 ^ y*2654435761 ^ z*805459861) & (2^19 - 1)
//
// Memory-bound gather kernel: 24 MB xyz stream + 64 MB L2-resident tables +
// 256 MB output stream. Strategy:
//   * one wave32 lane per (point, level) pair -> output stores are 256 B
//     contiguous per wave (perfectly coalesced b64 stores)
//   * non-temporal hints on the xyz/output streams so the 64 MB of hash
//     tables stay resident in the 192 MB L2 (gathers use default temporal)
//   * all hash math in uint32: (a*b) mod 2^19 only depends on low 32 bits,
//     and XOR commutes with the & mask, so this is bit-exact vs int64.
// ---------------------------------------------------------------------------

#define NUM_LEVELS   16
#define BASE_RES     16
#define LOG2_HASHMAP 19
#define TABLE_SIZE   (1u << LOG2_HASHMAP)
#define HASH_MASK    (TABLE_SIZE - 1u)
#define FEAT_DIM     2
#define PRIME1       2654435761u
#define PRIME2       805459861u

typedef float v2f __attribute__((ext_vector_type(2)));

__global__ __launch_bounds__(256)
void HashEncoding_11184094839334_kernel(const float* __restrict__ xyz,
                                        const float* __restrict__ tables,
                                        float* __restrict__ out,
                                        int n_points)
{
    // lane -> (point n, level l); consecutive lanes cover consecutive levels,
    // 16 lanes per point, 2 points per wave32.
    uint32_t tid = blockIdx.x * blockDim.x + threadIdx.x;
    uint32_t n = tid >> 4;
    uint32_t l = tid & 15u;
    if (n >= (uint32_t)n_points) return;

    // Streaming, read-once coordinates: non-temporal loads (TH=NT) so the
    // 24 MB xyz stream does not evict L2-resident hash tables. The 16 lanes
    // sharing a point issue identical addresses -> merged by the coalescer.
    const float* p = xyz + 3u * (size_t)n;
    float x = __builtin_nontemporal_load(p + 0);
    float y = __builtin_nontemporal_load(p + 1);
    float z = __builtin_nontemporal_load(p + 2);

    // Per-level resolution 16 * 2^l; f32 multiply + trunc-toward-zero matches
    // the reference's float32 scale followed by astype(int64) (coords >= 0).
    float res = (float)(BASE_RES << l);
    uint32_t ix = (uint32_t)(x * res);
    uint32_t iy = (uint32_t)(y * res);
    uint32_t iz = (uint32_t)(z * res);

    // Spatial hash, exact in 32-bit arithmetic (see header comment).
    uint32_t h = (ix ^ (iy * PRIME1) ^ (iz * PRIME2)) & HASH_MASK;

    // 8-byte gather from this level's table. Tables total 64 MB -> L2-hit
    // after warmup; keep default (regular temporal) caching policy.
    const v2f* tab = (const v2f*)tables;
    v2f feat = tab[(size_t)l * TABLE_SIZE + h];

    // Output [N, 32] f32 viewed as [N, 16] float2. Lanes of a wave write
    // 256 contiguous bytes -> one fully-coalesced NT b64 store each; NT keeps
    // the 256 MB output stream from rinsing the tables out of L2.
    v2f* o = (v2f*)out;
    __builtin_nontemporal_store(feat, o + ((size_t)n * NUM_LEVELS + l));
}

extern "C" void kernel_launch(void* const* d_in, const int* in_sizes, int n_in,
                              void* d_out, int out_size, void* d_ws, size_t ws_size,
                              hipStream_t stream) {
    const float* xyz    = (const float*)d_in[0];   // [N, 3] f32
    const float* tables = (const float*)d_in[1];   // [16, 2^19, 2] f32
    float* out = (float*)d_out;                    // [N, 32] f32

    int n_points = in_sizes[0] / 3;
    long long total = (long long)n_points * NUM_LEVELS;   // one lane per (n, l)
    int threads = 256;
    int blocks = (int)((total + threads - 1) / threads);

    HashEncoding_11184094839334_kernel<<<blocks, threads, 0, stream>>>(
        xyz, tables, out, n_points);
}